// GaussianRasterizer_59785944760976
// MI455X (gfx1250) — compile-verified
//
#include <hip/hip_runtime.h>
#include <hip/hip_bf16.h>
#include <math.h>

typedef float v2f __attribute__((ext_vector_type(2)));
typedef float v8f __attribute__((ext_vector_type(8)));

#define NMAX 1024
#define WID  256
#define HEI  256
#define HW   (WID * HEI)

// ---------------------------------------------------------------------------
// Kernel 1: per-gaussian preprocessing.
// Writes packed SoA (float4) arrays:
//   U0 = (mx, my, qa, qb)      quadratic-form coeffs: power = qa*dx^2+qb*dx*dy+qc*dy^2
//   U1 = (qc, opacity, depth, 0)
//   U2 = (xmin, xmax, ymin, ymax)
//   U3 = (r, g, b, 0)
// ---------------------------------------------------------------------------
__global__ __launch_bounds__(256) void gs_preprocess(
    const float* __restrict__ means3D, const float* __restrict__ colors,
    const float* __restrict__ opac,    const float* __restrict__ scales,
    const float* __restrict__ rots,    const float* __restrict__ viewm,
    const float* __restrict__ projm,   int N,
    float4* __restrict__ U0, float4* __restrict__ U1,
    float4* __restrict__ U2, float4* __restrict__ U3)
{
    int g = blockIdx.x * blockDim.x + threadIdx.x;
    if (g >= NMAX) return;
    if (g >= N) {                       // pad slots: never contribute, sort last
        U0[g] = make_float4(0.f, 0.f, 0.f, 0.f);
        U1[g] = make_float4(0.f, 0.f, 3.0e38f, 0.f);
        U2[g] = make_float4(0.f, 0.f, 0.f, 0.f);
        U3[g] = make_float4(0.f, 0.f, 0.f, 0.f);
        return;
    }
    float hx = means3D[3*g+0], hy = means3D[3*g+1], hz = means3D[3*g+2];

    // camera-space position (row i of viewmatrix dot homo)
    float cx = viewm[0]*hx + viewm[1]*hy + viewm[2]*hz  + viewm[3];
    float cy = viewm[4]*hx + viewm[5]*hy + viewm[6]*hz  + viewm[7];
    float cz = viewm[8]*hx + viewm[9]*hy + viewm[10]*hz + viewm[11];

    // clip / ndc / screen
    float clx = projm[0]*hx  + projm[1]*hy  + projm[2]*hz  + projm[3];
    float cly = projm[4]*hx  + projm[5]*hy  + projm[6]*hz  + projm[7];
    float clw = projm[12]*hx + projm[13]*hy + projm[14]*hz + projm[15];
    float iw  = 1.0f / (clw + 1e-7f);
    float mx  = ((clx*iw + 1.0f) * (float)WID - 1.0f) * 0.5f;
    float my  = ((cly*iw + 1.0f) * (float)HEI - 1.0f) * 0.5f;

    // rotation matrix from quaternion (r,x,y,z)
    float qr = rots[4*g+0], qx = rots[4*g+1], qy = rots[4*g+2], qz = rots[4*g+3];
    float R00 = 1.f-2.f*(qy*qy+qz*qz), R01 = 2.f*(qx*qy-qr*qz), R02 = 2.f*(qx*qz+qr*qy);
    float R10 = 2.f*(qx*qy+qr*qz), R11 = 1.f-2.f*(qx*qx+qz*qz), R12 = 2.f*(qy*qz-qr*qx);
    float R20 = 2.f*(qx*qz-qr*qy), R21 = 2.f*(qy*qz+qr*qx), R22 = 1.f-2.f*(qx*qx+qy*qy);
    float s0 = scales[3*g+0], s1 = scales[3*g+1], s2 = scales[3*g+2];
    float M00=R00*s0, M01=R01*s1, M02=R02*s2;
    float M10=R10*s0, M11=R11*s1, M12=R12*s2;
    float M20=R20*s0, M21=R21*s1, M22=R22*s2;

    // cov3D = M * M^T (symmetric)
    float S00 = M00*M00+M01*M01+M02*M02;
    float S01 = M00*M10+M01*M11+M02*M12;
    float S02 = M00*M20+M01*M21+M02*M22;
    float S11 = M10*M10+M11*M11+M12*M12;
    float S12 = M10*M20+M11*M21+M12*M22;
    float S22 = M20*M20+M21*M21+M22*M22;

    // perspective Jacobian (2x3)
    float fx  = projm[0] * (float)WID * 0.5f;
    float fy  = projm[5] * (float)HEI * 0.5f;
    float iz  = fx / (cz + 1e-7f);            // J00
    float izy = fy / (cz + 1e-7f);            // J11
    float J02 = -cx * fx / (cz*cz + 1e-7f);
    float J12 = -cy * fy / (cz*cz + 1e-7f);

    // cov2D = J * S * J^T + 0.3*I
    float T00 = iz*S00  + J02*S02;
    float T01 = iz*S01  + J02*S12;
    float T02 = iz*S02  + J02*S22;
    float T10 = izy*S01 + J12*S02;
    float T11 = izy*S11 + J12*S12;
    float T12 = izy*S12 + J12*S22;
    float c00 = T00*iz  + T02*J02 + 0.3f;
    float c01 = T01*izy + T02*J12;
    float c10 = T10*iz  + T12*J02;
    float c11 = T11*izy + T12*J12 + 0.3f;

    float det = c00*c11 - c01*c10;
    float sd  = (det < 1e-7f) ? 1.0f : det;
    float ia  =  c11 / sd;
    float ibc = (-c01 - c10) / sd;
    float id_ =  c00 / sd;
    float qa  = -0.5f * ia;
    float qb  = -0.5f * ibc;
    float qc  = -0.5f * id_;

    float sx3  = 3.0f * sqrtf(c00);
    float sy3  = 3.0f * sqrtf(c11);
    float xmin = floorf(fmaxf(0.0f, mx - sx3));
    float xmax = floorf(fminf((float)WID, mx + sx3 + 1.0f));
    float ymin = floorf(fmaxf(0.0f, my - sy3));
    float ymax = floorf(fminf((float)HEI, my + sy3 + 1.0f));
    bool valid = (mx >= -50.f) && (mx <= WID + 50.f) &&
                 (my >= -50.f) && (my <= HEI + 50.f) &&
                 (det >= 1e-7f) && (xmin < xmax) && (ymin < ymax);
    float op = valid ? opac[g] : 0.0f;
    if (!valid) { xmin = 0.f; xmax = 0.f; ymin = 0.f; ymax = 0.f; }

    U0[g] = make_float4(mx, my, qa, qb);
    U1[g] = make_float4(qc, op, cz, 0.f);
    U2[g] = make_float4(xmin, xmax, ymin, ymax);
    U3[g] = make_float4(colors[3*g+0], colors[3*g+1], colors[3*g+2], 0.f);
}

// ---------------------------------------------------------------------------
// Kernel 2: single-block LDS bitonic sort by depth (ascending = front-to-back)
// then gather the SoA into depth-sorted arrays S0..S3.
// ---------------------------------------------------------------------------
__global__ __launch_bounds__(512) void gs_sort_gather(
    const float4* __restrict__ U0, const float4* __restrict__ U1,
    const float4* __restrict__ U2, const float4* __restrict__ U3,
    float4* __restrict__ S0, float4* __restrict__ S1,
    float4* __restrict__ S2, float4* __restrict__ S3)
{
    __shared__ float key[NMAX];
    __shared__ int   idx[NMAX];
    int t = threadIdx.x;
    for (int s = 0; s < 2; ++s) {
        int i = t + s * 512;
        key[i] = U1[i].z;       // depth (padding slots hold +inf)
        idx[i] = i;
    }
    __syncthreads();
    for (int k = 2; k <= NMAX; k <<= 1) {
        for (int j = k >> 1; j > 0; j >>= 1) {
            for (int s = 0; s < 2; ++s) {
                int i   = t + s * 512;
                int ixj = i ^ j;
                if (ixj > i) {
                    bool up  = ((i & k) == 0);
                    float ki = key[i], kj = key[ixj];
                    if ((ki > kj) == up) {
                        key[i] = kj; key[ixj] = ki;
                        int tm = idx[i]; idx[i] = idx[ixj]; idx[ixj] = tm;
                    }
                }
            }
            __syncthreads();
        }
    }
    for (int s = 0; s < 2; ++s) {
        int i   = t + s * 512;
        int src = idx[i];
        S0[i] = U0[src]; S1[i] = U1[src]; S2[i] = U2[src]; S3[i] = U3[src];
    }
}

// ---------------------------------------------------------------------------
// Kernel 3: rasterize. 256 threads = 8 waves; wave w owns the 16-pixel row
// (blockIdx.x*16 .. +15, y = blockIdx.y*8 + w). Per 16-gaussian group one
// V_WMMA_F32_16X16X4_F32 evaluates power[g,p] (M=gaussian, N=pixel, K=3 terms
// [x'^2, x', 1] with x' tile-centered). Lanes 0-15 composite gaussians 0-7,
// lanes 16-31 gaussians 8-15; halves merged with a shfl_xor transmittance
// combine (C = C_lo + T_lo*C_hi, T = T_lo*T_hi). Gaussian data staged through
// LDS in 16 KB chunks, next chunk prefetched (global_prefetch_b8).
// ---------------------------------------------------------------------------
__global__ __launch_bounds__(256) void gs_raster(
    const float4* __restrict__ S0, const float4* __restrict__ S1,
    const float4* __restrict__ S2, const float4* __restrict__ S3,
    const float* __restrict__ bg, float* __restrict__ out)
{
    __shared__ float4 sG0[256], sG1[256], sG2[256], sG3[256];

    int tid  = threadIdx.x;
    int lane = tid & 31;
    int wave = tid >> 5;
    int lp   = lane & 15;                 // pixel slot within the row
    bool lo  = lane < 16;

    int   px  = blockIdx.x * 16 + lp;
    int   py  = blockIdx.y * 8 + wave;
    float pxf = (float)px;
    float pyf = (float)py;
    float xc  = (float)(blockIdx.x * 16) + 7.5f;   // tile-row center
    float xp  = (float)lp - 7.5f;                   // centered coordinate x'

    // B (K x N pixel geometry), loop-invariant:
    //   lanes 0-15 : (K0,K1) = (x'^2, x'); lanes 16-31: (K2,K3) = (1, 0)
    v2f B;
    B.x = lo ? xp * xp : 1.0f;
    B.y = lo ? xp      : 0.0f;

    float Rr = 0.f, Rg = 0.f, Rb = 0.f, Dacc = 0.f, T = 1.0f;

    for (int chunk = 0; chunk < NMAX; chunk += 256) {
        __syncthreads();
        int g = chunk + tid;
        sG0[tid] = S0[g]; sG1[tid] = S1[g]; sG2[tid] = S2[g]; sG3[tid] = S3[g];
        if (chunk + 256 < NMAX) {
            __builtin_prefetch(&S0[g + 256], 0, 0);
            __builtin_prefetch(&S1[g + 256], 0, 0);
            __builtin_prefetch(&S2[g + 256], 0, 0);
            __builtin_prefetch(&S3[g + 256], 0, 0);
        }
        __syncthreads();

        for (int grp = 0; grp < 256; grp += 16) {
            // A (M=gaussian x K): lane m<16 holds gaussian m's (K0,K1)=(qa,lin),
            // lane m+16 holds gaussian m's (K2,K3)=(const,0).
            float4 g0 = sG0[grp + lp];              // mx,my,qa,qb
            float4 g1 = sG1[grp + lp];              // qc,op,depth
            float ux  = xc  - g0.x;
            float uy  = pyf - g0.y;
            float qa  = g0.z, qb = g0.w, qc = g1.x;
            float lin = 2.0f * qa * ux + qb * uy;
            float cst = (qa * ux + qb * uy) * ux + qc * uy * uy;
            v2f A;
            A.x = lo ? qa  : cst;
            A.y = lo ? lin : 0.0f;

            v8f czero = {};
            // power[g,p] = qa*x'^2 + lin*x' + const  (f32 WMMA, EXEC all-ones)
            v8f pw = __builtin_amdgcn_wmma_f32_16x16x4_f32(
                false, A, false, B, (short)0, czero, false, false);

            // D layout: lanes 0-15 -> gaussians grp+0..7 (VGPR j = gaussian j),
            //           lanes 16-31 -> gaussians grp+8..15.
            int gb = grp + (lo ? 0 : 8);
            float lr = 0.f, lg = 0.f, lb = 0.f, ld = 0.f, lt = 1.0f;
            #pragma unroll
            for (int j = 0; j < 8; ++j) {
                float4 q1 = sG1[gb + j];
                float4 bb = sG2[gb + j];
                float4 cc = sG3[gb + j];
                float gs = __expf(fminf(pw[j], 0.0f));
                float a  = fminf(0.99f, q1.y * gs);
                bool in  = (pxf >= bb.x) && (pxf < bb.y) &&
                           (pyf >= bb.z) && (pyf < bb.w);
                a = in ? a : 0.0f;
                float w = a * lt;
                lr += cc.x * w; lg += cc.y * w; lb += cc.z * w; ld += q1.z * w;
                lt *= (1.0f - a);
            }

            // merge the two 8-gaussian half-scans across lane halves
            float orr = __shfl_xor(lr, 16, 32);
            float org = __shfl_xor(lg, 16, 32);
            float orb = __shfl_xor(lb, 16, 32);
            float ord = __shfl_xor(ld, 16, 32);
            float ort = __shfl_xor(lt, 16, 32);
            float CrLo = lo ? lr : orr, CrHi = lo ? orr : lr;
            float CgLo = lo ? lg : org, CgHi = lo ? org : lg;
            float CbLo = lo ? lb : orb, CbHi = lo ? orb : lb;
            float DLo  = lo ? ld : ord, DHi  = lo ? ord : ld;
            float TLo  = lo ? lt : ort, THi  = lo ? ort : lt;

            Rr   += T * (CrLo + TLo * CrHi);
            Rg   += T * (CgLo + TLo * CgHi);
            Rb   += T * (CbLo + TLo * CbHi);
            Dacc += T * (DLo  + TLo * DHi);
            T    *= TLo * THi;
        }
    }

    if (lo) {
        int pix = py * WID + px;
        out[0 * HW + pix] = bg[0] * T + Rr;
        out[1 * HW + pix] = bg[1] * T + Rg;
        out[2 * HW + pix] = bg[2] * T + Rb;
        out[3 * HW + pix] = Dacc;
        out[4 * HW + pix] = 1.0f - T;
    }
}

// ---------------------------------------------------------------------------
extern "C" void kernel_launch(void* const* d_in, const int* in_sizes, int n_in,
                              void* d_out, int out_size, void* d_ws, size_t ws_size,
                              hipStream_t stream)
{
    const float* means3D = (const float*)d_in[0];
    // d_in[1] = means2D (unused by the reference math)
    const float* colors  = (const float*)d_in[2];
    const float* opac    = (const float*)d_in[3];
    const float* scales  = (const float*)d_in[4];
    const float* rots    = (const float*)d_in[5];
    const float* bg      = (const float*)d_in[6];
    const float* viewm   = (const float*)d_in[7];
    const float* projm   = (const float*)d_in[8];
    int N = in_sizes[0] / 3;

    char* ws = (char*)d_ws;
    const size_t ARR = (size_t)NMAX * sizeof(float4);   // 16 KB each
    float4* S0 = (float4*)(ws + 0 * ARR);
    float4* S1 = (float4*)(ws + 1 * ARR);
    float4* S2 = (float4*)(ws + 2 * ARR);
    float4* S3 = (float4*)(ws + 3 * ARR);
    float4* U0 = (float4*)(ws + 4 * ARR);
    float4* U1 = (float4*)(ws + 5 * ARR);
    float4* U2 = (float4*)(ws + 6 * ARR);
    float4* U3 = (float4*)(ws + 7 * ARR);

    gs_preprocess<<<NMAX / 256, 256, 0, stream>>>(
        means3D, colors, opac, scales, rots, viewm, projm, N, U0, U1, U2, U3);

    gs_sort_gather<<<1, 512, 0, stream>>>(U0, U1, U2, U3, S0, S1, S2, S3);

    dim3 grid(WID / 16, HEI / 8);
    gs_raster<<<grid, 256, 0, stream>>>(S0, S1, S2, S3, bg, (float*)d_out);
}